// CenterTOp_5617817223882
// MI455X (gfx1250) — compile-verified
//
#include <hip/hip_runtime.h>
#include <math.h>

// ---------------------------------------------------------------------------
// Problem constants (fixed by the reference)
// ---------------------------------------------------------------------------
constexpr int NPTS = 128 * 128;   // 16384 points (FeatureT[0] flattened spatially)
constexpr int CDIM = 512;         // channels
constexpr int NB   = 4;           // batches
constexpr float THRV = 0.01f;

// ---------------------------------------------------------------------------
// Workspace layout (in 4-byte elements; float* and int* views share offsets)
// ---------------------------------------------------------------------------
constexpr int WS_NORM    = 0;                         // [NPTS]  per-point L2 norm (clamped to eps)
constexpr int WS_CN      = WS_NORM + NPTS;            // [2*CDIM] normalized centers (current)
constexpr int WS_CENTERS = WS_CN + 2 * CDIM;          // [2*CDIM] raw centers (current)
constexpr int WS_NEWSUM  = WS_CENTERS + 2 * CDIM;     // [2*CDIM] per-cluster channel sums
constexpr int WS_COUT    = WS_NEWSUM + 2 * CDIM;      // [2*CDIM] accumulated centers over batches
constexpr int WS_DCUR    = WS_COUT + 2 * CDIM;        // [NPTS*2] current distances d[n][k]
constexpr int WS_D2CALL  = WS_DCUR + 2 * NPTS;        // [NB*NPTS*2] snapshotted distances
constexpr int WS_RANGE   = WS_D2CALL + 2 * NB * NPTS; // [NB*2] (max-min+1e-7) per (b,k)
constexpr int WS_CD      = WS_RANGE + 8;              // scalar cd
constexpr int WS_CINI    = WS_CD + 1;                 // scalar cini accumulator
constexpr int WS_CI      = WS_CINI + 1;               // int: inner-step counter
constexpr int WS_LAB     = WS_CI + 1;                 // int[NPTS] current labels
constexpr int WS_LABALL  = WS_LAB + NPTS;             // int[NB*NPTS] snapshotted labels
// total ~1.07 MB

// ---------------------------------------------------------------------------
// Output layout (floats, concatenated in reference return order)
// ---------------------------------------------------------------------------
constexpr int OUT_C    = 0;                        // [2*CDIM]
constexpr int OUT_LAB  = 2 * CDIM;                 // [NB*NPTS]
constexpr int OUT_OH   = OUT_LAB + NB * NPTS;      // [NB*NPTS*2]
constexpr int OUT_W    = OUT_OH + NB * NPTS * 2;   // [NB*NPTS]
constexpr int OUT_CINI = OUT_W + NB * NPTS;        // scalar

typedef __attribute__((ext_vector_type(2))) float v2f;
typedef __attribute__((ext_vector_type(8))) float v8f;

__device__ inline bool step_inactive(const float* ws) {
  float cd = ws[WS_CD];
  int ci = ((const int*)ws)[WS_CI];
  return !(cd >= THRV && ci <= 3);
}

template <int BS>
__device__ inline float block_reduce_sum(float v, float* sm) {
  int t = threadIdx.x;
  sm[t] = v;
  __syncthreads();
  for (int off = BS / 2; off > 0; off >>= 1) {
    if (t < off) sm[t] += sm[t + off];
    __syncthreads();
  }
  float r = sm[0];
  __syncthreads();
  return r;
}

// ---------------------------------------------------------------------------
// K0: init centers (raw + normalized) and zero accumulators
// ---------------------------------------------------------------------------
__global__ void k_init(const float* __restrict__ centerInit, float* __restrict__ ws) {
  __shared__ float sm[256];
  int blk = blockIdx.x, t = threadIdx.x;
  if (blk < 2) {
    float s = 0.f;
    for (int c = t; c < CDIM; c += 256) {
      float x = centerInit[blk * CDIM + c];
      s = fmaf(x, x, s);
    }
    float nsq = block_reduce_sum<256>(s, sm);
    float nrm = fmaxf(sqrtf(nsq), 1e-12f);
    for (int c = t; c < CDIM; c += 256) {
      float x = centerInit[blk * CDIM + c];
      ws[WS_CENTERS + blk * CDIM + c] = x;
      ws[WS_CN + blk * CDIM + c] = x / nrm;
    }
  } else {
    for (int e = t; e < 2 * CDIM; e += 256) ws[WS_COUT + e] = 0.f;
    if (t == 0) {
      ws[WS_CD] = 0.f;
      ws[WS_CINI] = 0.f;
      ((int*)ws)[WS_CI] = 0;
    }
  }
}

// ---------------------------------------------------------------------------
// K1: per-point L2 norms of feats (feats[n][c] = F0[c*NPTS + n])
// ---------------------------------------------------------------------------
__global__ void k_norm(const float* __restrict__ F0, float* __restrict__ ws) {
  int n = blockIdx.x * blockDim.x + threadIdx.x;
  float s = 0.f;
  for (int c = 0; c < CDIM; ++c) {
    float x = F0[c * NPTS + n];
    s = fmaf(x, x, s);
  }
  ws[WS_NORM + n] = fmaxf(sqrtf(s), 1e-12f);
}

// ---------------------------------------------------------------------------
// S1: assignment via V_WMMA_F32_16X16X4_F32.
// Each wave: 16 points x 16 "centers" (2 real, 14 zero-padded), K = 512 in 4-chunks.
// A (16x4 f32, 2 VGPR): lanes 0-15 hold K=kb,kb+1 of row m; lanes 16-31 K=kb+2,kb+3.
// B (4x16 f32, 2 VGPR): lanes 0-15 rows K=0/1 of col j; lanes 16-31 rows K=2/3.
// B operand is pre-padded in LDS (sB[pair][col][2]) so every lane issues one
// unconditional aligned ds_load_b64 per chunk -> no EXEC-masked loads in the loop.
// ---------------------------------------------------------------------------
__global__ void k_assign(const float* __restrict__ F0, float* __restrict__ ws, int gated) {
  if (gated && step_inactive(ws)) return;

  __shared__ float sB[CDIM / 2][16][2];   // 32 KB: sB[p][j][q] = (j<2)?cn[j][2p+q]:0
  __shared__ float sdot[8][2][16];        // per-wave exchange of the 2 useful columns

  int t = threadIdx.x;
  for (int e = t; e < CDIM * 16; e += 256) {
    int p = e >> 5, j = (e >> 1) & 15, q = e & 1;
    ((float*)sB)[e] = (j < 2) ? ws[WS_CN + j * CDIM + (p * 2 + q)] : 0.f;
  }
  __syncthreads();

  int wave = t >> 5;
  int lane = t & 31;
  int m = lane & 15;
  int kb = (lane >> 4) << 1;                       // 0 for lanes 0-15, 2 for 16-31
  int pb = (blockIdx.x * 8 + wave) * 16;           // point tile base
  const float* colA = F0 + pb + m;
  const float* bptr = &sB[lane >> 4][m][0];        // folds the lane's K-half into base

  v8f acc = {};
  for (int c0 = 0; c0 < CDIM; c0 += 4) {
    v2f a;
    a.x = colA[(c0 + kb) * NPTS];
    a.y = colA[(c0 + kb + 1) * NPTS];
    v2f b = *(const v2f*)(bptr + (c0 >> 1) * 32);  // ds_load_b64, divergence-free
    acc = __builtin_amdgcn_wmma_f32_16x16x4_f32(false, a, false, b,
                                                (short)0, acc, false, false);
  }

  // D layout: VGPR r, lane L -> row M = r + 8*(L>=16), col = L&15.
  // Columns 0,1 (the two real centers) live in lanes {0,1,16,17}.
  if (m < 2) {
    int half = lane >> 4;
#pragma unroll
    for (int r = 0; r < 8; ++r) sdot[wave][m][half * 8 + r] = acc[r];
  }
  __syncthreads();

  if (lane < 16) {
    int n = pb + lane;
    float inv = 1.0f / ws[WS_NORM + n];
    float d0 = 0.5f * (1.0f - sdot[wave][0][lane] * inv);
    float d1 = 0.5f * (1.0f - sdot[wave][1][lane] * inv);
    int lab = (d1 < d0) ? 1 : 0;   // argmin, first-min tie-break
    ws[WS_DCUR + n * 2 + 0] = d0;
    ws[WS_DCUR + n * 2 + 1] = d1;
    ((int*)ws)[WS_LAB + n] = lab;
  }
}

// ---------------------------------------------------------------------------
// S2: per-channel segmented sums: new_sum[k][c] = sum_{lab[n]==k} feats[n][c]
// ---------------------------------------------------------------------------
__global__ void k_csum(const float* __restrict__ F0, float* __restrict__ ws, int gated) {
  if (gated && step_inactive(ws)) return;
  __shared__ float r0[256];
  __shared__ float r1[256];
  int c = blockIdx.x, t = threadIdx.x;
  const int* lab = (const int*)ws + WS_LAB;
  float s0 = 0.f, s1 = 0.f;
  for (int n = t; n < NPTS; n += 256) {
    float x = F0[c * NPTS + n];
    if (lab[n]) s1 += x; else s0 += x;
  }
  r0[t] = s0; r1[t] = s1;
  __syncthreads();
  for (int off = 128; off > 0; off >>= 1) {
    if (t < off) { r0[t] += r0[t + off]; r1[t] += r1[t + off]; }
    __syncthreads();
  }
  if (t == 0) {
    ws[WS_NEWSUM + c] = r0[0];
    ws[WS_NEWSUM + CDIM + c] = r1[0];
  }
}

// ---------------------------------------------------------------------------
// S3: counts, new centers, cosine-sim to old centers, state update. 1 block, 512 thr.
// ---------------------------------------------------------------------------
__global__ void k_update(float* __restrict__ ws, int gated) {
  if (gated && step_inactive(ws)) return;
  __shared__ float sm[512];
  int t = threadIdx.x;

  const int* lab = (const int*)ws + WS_LAB;
  float cs = 0.f;
  for (int n = t; n < NPTS; n += 512) cs += (float)lab[n];
  float count1 = block_reduce_sum<512>(cs, sm);
  float count0 = (float)NPTS - count1;

  float nc0 = ws[WS_NEWSUM + t] / (count0 + 1.0f);
  float nc1 = ws[WS_NEWSUM + CDIM + t] / (count1 + 1.0f);
  float cn0_old = ws[WS_CN + t];
  float cn1_old = ws[WS_CN + CDIM + t];

  float nsq0 = block_reduce_sum<512>(nc0 * nc0, sm);
  float nsq1 = block_reduce_sum<512>(nc1 * nc1, sm);
  float rn0 = fmaxf(sqrtf(nsq0), 1e-12f);
  float rn1 = fmaxf(sqrtf(nsq1), 1e-12f);
  float u0 = nc0 / rn0, u1 = nc1 / rn1;

  float sim0 = block_reduce_sum<512>(u0 * cn0_old, sm);
  float sim1 = block_reduce_sum<512>(u1 * cn1_old, sm);
  float cdn = 0.5f * (sim0 + sim1);

  ws[WS_CENTERS + t] = nc0;
  ws[WS_CENTERS + CDIM + t] = nc1;
  ws[WS_CN + t] = u0;
  ws[WS_CN + CDIM + t] = u1;

  if (t == 0) {
    int ci = ((int*)ws)[WS_CI];
    if (ci == 0) ws[WS_CINI] += cdn;
    ws[WS_CD] = cdn;
    ((int*)ws)[WS_CI] = ci + 1;
  }
}

// ---------------------------------------------------------------------------
// Batch boundary: snapshot (lab, oh, d) for batch b, accumulate cout, reset Ci.
// Always runs (matches reference batch_body).
// ---------------------------------------------------------------------------
__global__ void k_snap(float* __restrict__ ws, float* __restrict__ out, int b) {
  int n = blockIdx.x * 256 + threadIdx.x;
  int lab = ((const int*)ws)[WS_LAB + n];
  int bn = b * NPTS + n;
  out[OUT_LAB + bn] = (float)lab;
  out[OUT_OH + bn * 2 + 0] = lab ? 0.f : 1.f;
  out[OUT_OH + bn * 2 + 1] = lab ? 1.f : 0.f;
  ws[WS_D2CALL + bn * 2 + 0] = ws[WS_DCUR + n * 2 + 0];
  ws[WS_D2CALL + bn * 2 + 1] = ws[WS_DCUR + n * 2 + 1];
  ((int*)ws)[WS_LABALL + bn] = lab;
  if (blockIdx.x == 0) {
    for (int e = threadIdx.x; e < 2 * CDIM; e += 256)
      ws[WS_COUT + e] += ws[WS_CENTERS + e];
    if (threadIdx.x == 0) ((int*)ws)[WS_CI] = 0;
  }
}

// ---------------------------------------------------------------------------
// F1: per-(batch, column) range over N: max - min + 1e-7
// ---------------------------------------------------------------------------
__global__ void k_range(float* __restrict__ ws) {
  __shared__ float smx[256];
  __shared__ float smn[256];
  int bk = blockIdx.x;           // b*2 + k
  int b = bk >> 1, k = bk & 1;
  int t = threadIdx.x;
  float mx = -3.4e38f, mn = 3.4e38f;
  for (int n = t; n < NPTS; n += 256) {
    float v = ws[WS_D2CALL + (b * NPTS + n) * 2 + k];
    mx = fmaxf(mx, v);
    mn = fminf(mn, v);
  }
  smx[t] = mx; smn[t] = mn;
  __syncthreads();
  for (int off = 128; off > 0; off >>= 1) {
    if (t < off) {
      smx[t] = fmaxf(smx[t], smx[t + off]);
      smn[t] = fminf(smn[t], smn[t + off]);
    }
    __syncthreads();
  }
  if (t == 0) ws[WS_RANGE + bk] = smx[0] - smn[0] + 1e-7f;
}

// ---------------------------------------------------------------------------
// F2: weight[b,n] = 1.1 + d[b,n,lab] / range[b,lab]
//   (d' = -d/range; d'' = 1 - d' + 0.1; weight = sum_k d''*oh = 1.1 + d/range at lab)
// ---------------------------------------------------------------------------
__global__ void k_weight(const float* __restrict__ ws, float* __restrict__ out) {
  int bn = blockIdx.x * 256 + threadIdx.x;   // 0..NB*NPTS-1
  int b = bn >> 14;                          // NPTS = 2^14
  int lab = ((const int*)ws)[WS_LABALL + bn];
  float v = ws[WS_D2CALL + bn * 2 + lab];
  out[OUT_W + bn] = 1.1f + v / ws[WS_RANGE + b * 2 + lab];
}

// ---------------------------------------------------------------------------
// F3: centersIterout = cout/B ; cinidist = cini/B
// ---------------------------------------------------------------------------
__global__ void k_final(const float* __restrict__ ws, float* __restrict__ out) {
  for (int e = threadIdx.x; e < 2 * CDIM; e += 256)
    out[OUT_C + e] = ws[WS_COUT + e] * 0.25f;
  if (threadIdx.x == 0) out[OUT_CINI] = ws[WS_CINI] * 0.25f;
}

// ---------------------------------------------------------------------------
// Host-side sequence (graph-capturable: fixed kernel list, data-dependent
// masking handled inside gated kernels via the cd/Ci flags in workspace).
// ---------------------------------------------------------------------------
extern "C" void kernel_launch(void* const* d_in, const int* in_sizes, int n_in,
                              void* d_out, int out_size, void* d_ws, size_t ws_size,
                              hipStream_t stream) {
  (void)in_sizes; (void)n_in; (void)out_size; (void)ws_size;

  const float* FeatureT  = (const float*)d_in[0];
  const float* centerInit = (const float*)d_in[1];
  const float* F0 = FeatureT;                 // bug-faithful: only batch 0 is clustered
  float* ws  = (float*)d_ws;
  float* out = (float*)d_out;

  k_init<<<3, 256, 0, stream>>>(centerInit, ws);
  k_norm<<<NPTS / 256, 256, 0, stream>>>(F0, ws);

  auto run_step = [&](int gated) {
    k_assign<<<NPTS / (16 * 8), 256, 0, stream>>>(F0, ws, gated);  // 128 blocks
    k_csum<<<CDIM, 256, 0, stream>>>(F0, ws, gated);
    k_update<<<1, 512, 0, stream>>>(ws, gated);
  };

  run_step(0);                                // unconditional first step
  for (int b = 0; b < NB; ++b) {
    for (int i = 0; i < 4; ++i) run_step(1);  // gated on (cd >= THR && Ci <= 3)
    k_snap<<<NPTS / 256, 256, 0, stream>>>(ws, out, b);
  }

  k_range<<<NB * 2, 256, 0, stream>>>(ws);
  k_weight<<<NB * NPTS / 256, 256, 0, stream>>>(ws, out);
  k_final<<<1, 256, 0, stream>>>(ws, out);
}